// GCN_53352083751031
// MI455X (gfx1250) — compile-verified
//
#include <hip/hip_runtime.h>

typedef __attribute__((ext_vector_type(2))) float v2f;
typedef __attribute__((ext_vector_type(8))) float v8f;

#define BN_EPS 1e-5f

// ---------------------------------------------------------------------------
// Utility fills
// ---------------------------------------------------------------------------
__global__ void fill_kernel(float* __restrict__ p, float val, long long n) {
    long long i = (long long)blockIdx.x * blockDim.x + threadIdx.x;
    if (i < n) p[i] = val;
}

// deg[dst[e]] += 1  (edges only; self-loop folded into init value 1.0)
__global__ void deg_accum_kernel(const long long* __restrict__ dst,
                                 float* __restrict__ deg, long long E) {
    long long e = (long long)blockIdx.x * blockDim.x + threadIdx.x;
    if (e < E) atomicAdd(&deg[(long long)dst[e]], 1.0f);
}

// deg -> D^-1/2
__global__ void dinv_kernel(float* __restrict__ deg, int n) {
    int i = blockIdx.x * blockDim.x + threadIdx.x;
    if (i < n) {
        float d = deg[i];
        deg[i] = d > 0.0f ? rsqrtf(d) : 0.0f;
    }
}

// ---------------------------------------------------------------------------
// Dense GEMM: O[nrows x 64] = X[nrows x K] @ W[K x 64] via V_WMMA_F32_16X16X4_F32
// One wave = one 16-row tile, all four 16-col tiles (4 x v8f accumulators).
// W is staged in LDS pre-swizzled so each lane's B fragment (b.x,b.y) is a
// single aligned 8-byte ds_load_b64 (no VGPR repacking movs).
//   pair index p = (k/4)*128 + ct*32 + lane
//   sW[2p+j] = W[k + 2*(lane>>4) + j][ct*16 + (lane&15)]
// ---------------------------------------------------------------------------
template <int K>
__global__ void gemm_wmma_kernel(const float* __restrict__ X,
                                 const float* __restrict__ W,
                                 float* __restrict__ O, int nrows) {
    __shared__ __align__(16) float sW[K * 64];
    for (int p = threadIdx.x; p < K * 32; p += blockDim.x) {
        int l    = p & 31;
        int ct   = (p >> 5) & 3;
        int kq   = p >> 7;
        int row0 = kq * 4 + ((l >> 4) << 1);
        int col  = ct * 16 + (l & 15);
        sW[p * 2 + 0] = W[row0 * 64 + col];
        sW[p * 2 + 1] = W[(row0 + 1) * 64 + col];
    }
    __syncthreads();

    const int wave = threadIdx.x >> 5;       // 8 waves / block
    const int lane = threadIdx.x & 31;
    const int rowBase = (blockIdx.x * 8 + wave) * 16;
    if (rowBase >= nrows) return;            // wave-uniform: nrows % 16 == 0

    const int m  = lane & 15;                // row within tile (A)
    const int kh = (lane >> 4) << 1;         // K sub-pair: 0 or 2

    v8f acc[4] = {};
    const float* xrow = X + (long long)(rowBase + m) * K;

    for (int k = 0; k < K; k += 4) {
        // A fragment: lane<16 holds K=k,k+1 ; lane>=16 holds K=k+2,k+3
        float2 av = *(const float2*)(xrow + k + kh);
        v2f a; a.x = av.x; a.y = av.y;
        const float* bbase = sW + (((k >> 2) * 128 + lane) << 1);
#pragma unroll
        for (int ct = 0; ct < 4; ++ct) {
            v2f b = *(const v2f*)(bbase + ct * 64);   // ds_load_b64
            acc[ct] = __builtin_amdgcn_wmma_f32_16x16x4_f32(
                false, a, false, b, (short)0, acc[ct], false, false);
        }
    }

    // D layout: VGPR v, lane<16 -> M=v ; lane>=16 -> M=v+8 ; N = lane&15
    const int moff = (lane >> 4) << 3;
#pragma unroll
    for (int ct = 0; ct < 4; ++ct) {
#pragma unroll
        for (int v = 0; v < 8; ++v) {
            int row = rowBase + v + moff;
            O[(long long)row * 64 + ct * 16 + m] = acc[ct][v];
        }
    }
}

// ---------------------------------------------------------------------------
// agg[i,:] = h[i,:] * dinv[i]^2 + bias   (self-loop term; also initializes agg)
// 16 threads per node, float4 per thread.
// ---------------------------------------------------------------------------
__global__ void self_bias_kernel(const float* __restrict__ h,
                                 const float* __restrict__ dinv,
                                 const float* __restrict__ bias,
                                 float* __restrict__ agg, int n) {
    long long t = (long long)blockIdx.x * blockDim.x + threadIdx.x;
    long long i = t >> 4;
    if (i >= n) return;
    int f4 = (int)(t & 15) * 4;
    float di = dinv[i];
    float w = di * di;
    float4 hv = *(const float4*)(h + i * 64 + f4);
    float4 bv = *(const float4*)(bias + f4);
    float4 o;
    o.x = hv.x * w + bv.x;
    o.y = hv.y * w + bv.y;
    o.z = hv.z * w + bv.z;
    o.w = hv.w * w + bv.w;
    *(float4*)(agg + i * 64 + f4) = o;
}

// ---------------------------------------------------------------------------
// Edge scatter: agg[dst,:] += h[src,:] * dinv[src]*dinv[dst]
// 16 threads per edge, float4 gather + 4 float atomics.
// ---------------------------------------------------------------------------
__global__ void scatter_kernel(const long long* __restrict__ src,
                               const long long* __restrict__ dst,
                               const float* __restrict__ dinv,
                               const float* __restrict__ h,
                               float* __restrict__ agg, long long E) {
    long long t = (long long)blockIdx.x * blockDim.x + threadIdx.x;
    long long e = t >> 4;
    if (e >= E) return;
    int f4 = (int)(t & 15) * 4;
    long long s = src[e];
    long long d = dst[e];
    float w = dinv[s] * dinv[d];
    float4 hv = *(const float4*)(h + s * 64 + f4);
    float* p = agg + d * 64 + f4;
    atomicAdd(p + 0, hv.x * w);
    atomicAdd(p + 1, hv.y * w);
    atomicAdd(p + 2, hv.z * w);
    atomicAdd(p + 3, hv.w * w);
}

// ---------------------------------------------------------------------------
// BatchNorm statistics: per-feature sum / sumsq into stats[0..63], stats[64..127]
// ---------------------------------------------------------------------------
__global__ void bn_stats_kernel(const float* __restrict__ agg,
                                float* __restrict__ stats, int n) {
    int f = threadIdx.x & 63;
    int r = blockIdx.x * (blockDim.x >> 6) + (threadIdx.x >> 6);
    int stride = gridDim.x * (blockDim.x >> 6);
    float s = 0.0f, sq = 0.0f;
    for (; r < n; r += stride) {
        float v = agg[(long long)r * 64 + f];
        s += v;
        sq += v * v;
    }
    atomicAdd(&stats[f], s);
    atomicAdd(&stats[64 + f], sq);
}

// scale/shift from accumulated stats: stats[128+f]=scale, stats[192+f]=shift
__global__ void bn_finalize_kernel(float* __restrict__ stats,
                                   const float* __restrict__ gamma,
                                   const float* __restrict__ beta, int n) {
    int f = threadIdx.x;
    if (f < 64) {
        float inv_n = 1.0f / (float)n;
        float mean = stats[f] * inv_n;
        float var = stats[64 + f] * inv_n - mean * mean;
        float scale = gamma[f] * rsqrtf(var + BN_EPS);
        stats[128 + f] = scale;
        stats[192 + f] = beta[f] - mean * scale;
    }
}

// In-place BN apply + ReLU (float4 over N*64)
__global__ void bn_apply_relu_kernel(float* __restrict__ agg,
                                     const float* __restrict__ stats, int n) {
    long long t = (long long)blockIdx.x * blockDim.x + threadIdx.x;
    long long i = t >> 4;
    if (i >= n) return;
    int f4 = (int)(t & 15) * 4;
    float4 v = *(const float4*)(agg + i * 64 + f4);
    float4 sc = *(const float4*)(stats + 128 + f4);
    float4 sh = *(const float4*)(stats + 192 + f4);
    float4 o;
    o.x = fmaxf(v.x * sc.x + sh.x, 0.0f);
    o.y = fmaxf(v.y * sc.y + sh.y, 0.0f);
    o.z = fmaxf(v.z * sc.z + sh.z, 0.0f);
    o.w = fmaxf(v.w * sc.w + sh.w, 0.0f);
    *(float4*)(agg + i * 64 + f4) = o;
}

// ---------------------------------------------------------------------------
// Launch
// ---------------------------------------------------------------------------
extern "C" void kernel_launch(void* const* d_in, const int* in_sizes, int n_in,
                              void* d_out, int out_size, void* d_ws, size_t ws_size,
                              hipStream_t stream) {
    const float* x = (const float*)d_in[0];
    const long long* ei = (const long long*)d_in[1];   // int64 edge_index [2, E]
    const float* W1 = (const float*)d_in[2];
    const float* b1 = (const float*)d_in[3];
    const float* W2 = (const float*)d_in[4];
    const float* b2 = (const float*)d_in[5];
    const float* gamma = (const float*)d_in[6];
    const float* beta = (const float*)d_in[7];
    float* out = (float*)d_out;

    const int N = in_sizes[0] / 128;          // F_in = 128
    const long long E = (long long)in_sizes[1] / 2;
    const long long* src = ei;
    const long long* dst = ei + E;

    // Workspace layout (floats), all offsets kept 16B-aligned
    float* ws = (float*)d_ws;
    long long degN = ((long long)N + 63) & ~63LL;
    float* dinv = ws;                          // N (deg, then D^-1/2 in place)
    float* stats = ws + degN;                  // 256: sum, sumsq, scale, shift
    float* bufA = stats + 256;                 // N*64: h = X@W (both layers)
    float* bufB = bufA + (long long)N * 64;    // N*64: aggregated/BN activations

    const int B = 256;
    dim3 blk(B);

    // --- degrees ---
    fill_kernel<<<dim3((N + B - 1) / B), blk, 0, stream>>>(dinv, 1.0f, N); // self-loops
    fill_kernel<<<dim3(1), blk, 0, stream>>>(stats, 0.0f, 128);
    deg_accum_kernel<<<dim3((unsigned)((E + B - 1) / B)), blk, 0, stream>>>(dst, dinv, E);
    dinv_kernel<<<dim3((N + B - 1) / B), blk, 0, stream>>>(dinv, N);

    // --- layer 1: h1 = x @ W1 (fp32 WMMA) ---
    int gemmBlocks = (N + 127) / 128;          // 8 waves * 16 rows per block
    gemm_wmma_kernel<128><<<dim3(gemmBlocks), blk, 0, stream>>>(x, W1, bufA, N);

    // --- aggregate: agg = D^-1/2 (A+I) D^-1/2 h1 + b1 ---
    long long nodeT = (long long)N * 16;
    long long edgeT = E * 16;
    self_bias_kernel<<<dim3((unsigned)((nodeT + B - 1) / B)), blk, 0, stream>>>(
        bufA, dinv, b1, bufB, N);
    scatter_kernel<<<dim3((unsigned)((edgeT + B - 1) / B)), blk, 0, stream>>>(
        src, dst, dinv, bufA, bufB, E);

    // --- BatchNorm (batch stats) + ReLU, in place on bufB ---
    bn_stats_kernel<<<dim3(512), blk, 0, stream>>>(bufB, stats, N);
    bn_finalize_kernel<<<dim3(1), dim3(64), 0, stream>>>(stats, gamma, beta, N);
    bn_apply_relu_kernel<<<dim3((unsigned)((nodeT + B - 1) / B)), blk, 0, stream>>>(
        bufB, stats, N);

    // --- layer 2: h2 = hbn @ W2 (fp32 WMMA), aggregate into d_out ---
    gemm_wmma_kernel<64><<<dim3(gemmBlocks), blk, 0, stream>>>(bufB, W2, bufA, N);
    self_bias_kernel<<<dim3((unsigned)((nodeT + B - 1) / B)), blk, 0, stream>>>(
        bufA, dinv, b2, out, N);
    scatter_kernel<<<dim3((unsigned)((edgeT + B - 1) / B)), blk, 0, stream>>>(
        src, dst, dinv, bufA, out, E);
}